// TopKTopPSampler_13640816132647
// MI455X (gfx1250) — compile-verified
//
#include <hip/hip_runtime.h>
#include <hip/hip_bf16.h>
#include <math.h>
#include <stdint.h>

// One workgroup (1024 threads = 32 wave32) per row of [B=256, V=128000].
// Memory-bound sampler: async global->LDS DMA streaming, LDS histograms,
// non-temporal IO so logits stay L2-resident across the re-read passes.

#define BDIM   1024
#define NBINS  4096
#define NCHUNK (NBINS / 32)

typedef __attribute__((ext_vector_type(4))) float fvec4;

__device__ __forceinline__ unsigned ordkey(float x) {
  unsigned u = __float_as_uint(x);
  return (u & 0x80000000u) ? ~u : (u | 0x80000000u);
}
__device__ __forceinline__ float ordinv(unsigned k) {
  unsigned u = (k & 0x80000000u) ? (k & 0x7fffffffu) : ~k;
  return __uint_as_float(u);
}

// --- CDNA5 async global->LDS DMA (ASYNCcnt-tracked) ------------------------
__device__ __forceinline__ void async_load_b128_to_lds(const fvec4* gsrc,
                                                       unsigned lds_byte_off) {
  asm volatile("global_load_async_to_lds_b128 %0, %1, off"
               :: "v"(lds_byte_off), "v"(gsrc)
               : "memory");
}
__device__ __forceinline__ void s_wait_async0() {
  asm volatile("s_wait_asynccnt 0x0" ::: "memory");
}
__device__ __forceinline__ void s_wait_async1() {
  asm volatile("s_wait_asynccnt 0x1" ::: "memory");
}

__global__ __launch_bounds__(BDIM) void topk_topp_sample_kernel(
    const float* __restrict__ logits,
    const int*   __restrict__ karr,
    const float* __restrict__ parr,
    const float* __restrict__ qarr,
    float* __restrict__ out_ids,
    float* __restrict__ out_logprobs,
    int V)
{
  __shared__ unsigned s_hist[NBINS];      // counts, then reused as f32 exp-sums
  __shared__ unsigned s_chunkU[NCHUNK];
  __shared__ float    s_chunkF[NCHUNK];
  __shared__ unsigned s_maxkey;
  __shared__ unsigned s_thrK;
  __shared__ unsigned s_thr;
  __shared__ float    s_M;
  __shared__ float    s_S;
  __shared__ float    s_logS;
  __shared__ unsigned long long s_arg;
  __shared__ float    s_stage[2][BDIM * 4];   // 2 x 16KB async staging buffers

  float* s_histf = (float*)s_hist;
  const int row = blockIdx.x;
  const int tid = threadIdx.x;
  const float    p  = parr[row];
  const unsigned kk = (unsigned)karr[row];
  const fvec4* xp = (const fvec4*)(logits + (size_t)row * V);
  const fvec4* qp = (const fvec4*)(qarr  + (size_t)row * V);
  fvec4*       op = (fvec4*)(out_logprobs + (size_t)row * V);
  const int nvec   = V >> 2;
  const int ntiles = (nvec + BDIM - 1) / BDIM;
  const float NEGINF = __uint_as_float(0xff800000u);

  for (int i = tid; i < NBINS; i += BDIM) s_hist[i] = 0u;
  if (tid == 0) { s_maxkey = 0u; s_S = 0.f; s_arg = 0ull; }
  __syncthreads();

  // ---- Pass 1: key histogram + exact max, async double-buffered staging ---
  {
    const unsigned lds0 = (unsigned)(uintptr_t)(&s_stage[0][tid * 4]);
    const unsigned lds1 = (unsigned)(uintptr_t)(&s_stage[1][tid * 4]);
    {
      int i0 = tid; if (i0 > nvec - 1) i0 = nvec - 1;   // uniform issue
      async_load_b128_to_lds(xp + i0, lds0);
    }
    unsigned lmax = 0u;
    for (int t = 0; t < ntiles; ++t) {
      if (t + 1 < ntiles) {
        int in = (t + 1) * BDIM + tid; if (in > nvec - 1) in = nvec - 1;
        async_load_b128_to_lds(xp + in, ((t + 1) & 1) ? lds1 : lds0);
        s_wait_async1();   // tile t complete; tile t+1 may remain in flight
      } else {
        s_wait_async0();
      }
      const int i = t * BDIM + tid;
      if (i < nvec) {
        const float* sb = &s_stage[t & 1][tid * 4];
        #pragma unroll
        for (int c = 0; c < 4; ++c) {
          unsigned key = ordkey(sb[c]);
          atomicAdd(&s_hist[key >> 20], 1u);        // ds_add_u32
          if (key > lmax) lmax = key;
        }
      }
    }
    atomicMax(&s_maxkey, lmax);                      // ds_max_u32
  }
  __syncthreads();

  if (tid < NCHUNK) {
    unsigned s = 0;
    #pragma unroll
    for (int j = 0; j < 32; ++j) s += s_hist[tid * 32 + j];
    s_chunkU[tid] = s;
  }
  __syncthreads();

  // ---- top-k threshold: first bin (from top) where cum count reaches k ----
  if (tid == 0) {
    unsigned cum = 0; int thrbin = 0; bool found = false;
    for (int ch = NCHUNK - 1; ch >= 0 && !found; --ch) {
      unsigned cs = s_chunkU[ch];
      if (cum + cs < kk) { cum += cs; continue; }
      for (int b = ch * 32 + 31; b >= ch * 32; --b) {
        if (cum + s_hist[b] >= kk) { thrbin = b; found = true; break; }
        cum += s_hist[b];
      }
    }
    s_thrK = (unsigned)thrbin << 20;
    s_M = ordinv(s_maxkey);
  }
  __syncthreads();

  for (int i = tid; i < NBINS; i += BDIM) s_hist[i] = 0u;   // re-zero (== 0.0f)
  __syncthreads();

  // ---- Pass 2: exp-sum histogram over top-k survivors (L2-resident row) ---
  {
    const float M = s_M; const unsigned thrK = s_thrK;
    for (int i = tid; i < nvec; i += BDIM) {
      fvec4 xv = xp[i];
      #pragma unroll
      for (int c = 0; c < 4; ++c) {
        float v = xv[c];
        unsigned key = ordkey(v);
        if (key >= thrK) atomicAdd(&s_histf[key >> 20], __expf(v - M)); // ds_add_f32
      }
    }
  }
  __syncthreads();

  if (tid < NCHUNK) {
    float s = 0.f;
    #pragma unroll
    for (int j = 0; j < 32; ++j) s += s_histf[tid * 32 + j];
    s_chunkF[tid] = s;
  }
  __syncthreads();

  // ---- top-p threshold: lowest bin whose mass-from-top stays < p*S --------
  if (tid == 0) {
    float S = 0.f;
    for (int ch = 0; ch < NCHUNK; ++ch) S += s_chunkF[ch];
    const float target = p * S;
    float cum = 0.f; int thrbin = 0; bool found = false;
    for (int ch = NCHUNK - 1; ch >= 0 && !found; --ch) {
      float cs = s_chunkF[ch];
      if (cum + cs < target) { cum += cs; continue; }
      for (int b = ch * 32 + 31; b >= ch * 32; --b) {
        if (cum >= target) { thrbin = b + 1; found = true; break; }
        cum += s_histf[b];
      }
    }
    unsigned thrP = (unsigned)thrbin << 20;
    s_thr = (s_thrK > thrP) ? s_thrK : thrP;
  }
  __syncthreads();

  // ---- Pass 3: exact masked exp-sum + Gumbel-max argmax(x - log q) --------
  {
    const float M = s_M; const unsigned thr = s_thr;
    float localS = 0.f;
    unsigned long long best = 0ull;
    for (int i = tid; i < nvec; i += BDIM) {
      fvec4 xv = xp[i];
      fvec4 qv = __builtin_nontemporal_load(qp + i);   // q is single-use
      #pragma unroll
      for (int c = 0; c < 4; ++c) {
        float v = xv[c];
        unsigned key = ordkey(v);
        if (key >= thr) {
          localS += __expf(v - M);
          float sc = v - __logf(qv[c]);                // monotone in probs/q
          unsigned idx = (unsigned)(4 * i + c);
          unsigned long long pk =
              ((unsigned long long)ordkey(sc) << 32) | (unsigned)(~idx);
          if (pk > best) best = pk;                    // tie -> smallest idx
        }
      }
    }
    #pragma unroll
    for (int o = 16; o > 0; o >>= 1) localS += __shfl_xor(localS, o, 32);
    if ((tid & 31) == 0) atomicAdd(&s_S, localS);
    atomicMax(&s_arg, best);                           // ds_max_u64
  }
  __syncthreads();

  if (tid == 0) {
    s_logS = __logf(s_S);
    out_ids[row] = (float)(~(unsigned)s_arg);
  }
  __syncthreads();

  // ---- Pass 4: log_softmax of masked logits, non-temporal stores ----------
  {
    const float M = s_M, lS = s_logS; const unsigned thr = s_thr;
    for (int i = tid; i < nvec; i += BDIM) {
      fvec4 xv = xp[i];
      fvec4 lp;
      #pragma unroll
      for (int c = 0; c < 4; ++c)
        lp[c] = (ordkey(xv[c]) >= thr) ? (xv[c] - M - lS) : NEGINF;
      __builtin_nontemporal_store(lp, op + i);
    }
  }
}

extern "C" void kernel_launch(void* const* d_in, const int* in_sizes, int n_in,
                              void* d_out, int out_size, void* d_ws, size_t ws_size,
                              hipStream_t stream) {
  const float* logits = (const float*)d_in[0];
  const int*   karr   = (const int*)d_in[1];
  const float* parr   = (const float*)d_in[2];
  const float* qarr   = (const float*)d_in[3];
  const int B = in_sizes[1];                       // k has B entries
  const int V = (int)((long long)in_sizes[0] / B); // logits is B*V
  float* out_ids = (float*)d_out;                  // tuple out #0: ids (B)
  float* out_lp  = out_ids + B;                    // tuple out #1: logprobs (B*V)
  topk_topp_sample_kernel<<<B, BDIM, 0, stream>>>(logits, karr, parr, qarr,
                                                  out_ids, out_lp, V);
}